// NCP_63350767616284
// MI455X (gfx1250) — compile-verified
//
#include <hip/hip_runtime.h>
#include <hip/hip_bf16.h>
#include <math.h>

#define T_LEN   512
#define D_DIM   512
#define U_DIM   128
#define M_DIM   46
#define G_DIM   512   // 4*U
#define UNFOLDS 6
#define EPS_LTC 1e-8f

typedef __attribute__((ext_vector_type(2))) float v2f;
typedef __attribute__((ext_vector_type(8))) float v8f;

__device__ __forceinline__ float sigmoidf_(float x) {
  return 1.0f / (1.0f + __expf(-x));
}
__device__ __forceinline__ float softplusf_(float x) {
  return (x > 15.0f) ? x : log1pf(expf(x));
}

// ---------------------------------------------------------------------------
// Kernel 1: gather embeddings X[t, :] = embed_table[sentence[t], :]
// ---------------------------------------------------------------------------
__global__ void ncp_gather_embed(const int* __restrict__ sentence,
                                 const float* __restrict__ table,
                                 float* __restrict__ X) {
  int t = blockIdx.x;
  long long row = (long long)sentence[t] * D_DIM;
  for (int d = threadIdx.x; d < D_DIM; d += blockDim.x)
    X[t * D_DIM + d] = table[row + d];
}

// ---------------------------------------------------------------------------
// Kernel 2: precompute softplus params, packed as float4 per synapse:
//   sensp[d*U+u] = {softplus(sens_w)*sens_mask, sens_mu, sens_sigma, sens_erev}
//   synp [s*U+u] = {softplus(w)*mask,           mu,      sigma,      erev}
//   cmt = softplus(cm)*UNFOLDS, gl = softplus(gleak)
// ---------------------------------------------------------------------------
__global__ void ncp_precompute(const float* __restrict__ sens_w,
                               const float* __restrict__ sens_mask,
                               const float* __restrict__ sens_mu,
                               const float* __restrict__ sens_sigma,
                               const float* __restrict__ sens_erev,
                               const float* __restrict__ w,
                               const float* __restrict__ mask,
                               const float* __restrict__ mu,
                               const float* __restrict__ sigma,
                               const float* __restrict__ erev,
                               const float* __restrict__ cm,
                               const float* __restrict__ gleak,
                               float4* __restrict__ sensp,
                               float4* __restrict__ synp,
                               float* __restrict__ cmt,
                               float* __restrict__ gl) {
  int i = blockIdx.x * blockDim.x + threadIdx.x;
  if (i < D_DIM * U_DIM) {
    float sw = softplusf_(sens_w[i]) * sens_mask[i];
    sensp[i] = make_float4(sw, sens_mu[i], sens_sigma[i], sens_erev[i]);
  }
  if (i < U_DIM * U_DIM) {
    float wv = softplusf_(w[i]) * mask[i];
    synp[i] = make_float4(wv, mu[i], sigma[i], erev[i]);
  }
  if (i < U_DIM) {
    cmt[i] = softplusf_(cm[i]) * (float)UNFOLDS;
    gl[i]  = softplusf_(gleak[i]);
  }
}

// ---------------------------------------------------------------------------
// Kernel 3: WMMA GEMM  Z = X @ W_ih + b_ih   (512x512x512, fp32)
// One wave per 16x16 tile; K marched in steps of 4 with v_wmma_f32_16x16x4_f32.
// ---------------------------------------------------------------------------
__global__ void ncp_gemm_wmma(const float* __restrict__ X,
                              const float* __restrict__ Wih,
                              const float* __restrict__ bih,
                              float* __restrict__ Z) {
  const int lane  = threadIdx.x & 31;
  const int wave  = threadIdx.x >> 5;
  const int tile  = blockIdx.x * 8 + wave;       // 1024 tiles total
  const int tm    = tile >> 5;                   // row tile (0..31)
  const int tn    = tile & 31;                   // col tile (0..31)
  const int half  = lane >> 4;
  const int lm    = lane & 15;
  const int arow  = tm * 16 + lm;
  const int bcol  = tn * 16 + lm;

  v8f acc = {};
#if defined(__HIP_DEVICE_COMPILE__) && __has_builtin(__builtin_amdgcn_wmma_f32_16x16x4_f32)
  #pragma unroll 4
  for (int k0 = 0; k0 < D_DIM; k0 += 4) {
    const int ka = k0 + half * 2;
    v2f a, b;
    a.x = X[arow * D_DIM + ka];
    a.y = X[arow * D_DIM + ka + 1];
    b.x = Wih[ka * G_DIM + bcol];
    b.y = Wih[(ka + 1) * G_DIM + bcol];
    acc = __builtin_amdgcn_wmma_f32_16x16x4_f32(
        /*neg_a=*/false, a, /*neg_b=*/false, b,
        /*c_mod=*/(short)0, acc, /*reuse_a=*/false, /*reuse_b=*/false);
  }
#else
  for (int k = 0; k < D_DIM; ++k) {
    float bv = Wih[k * G_DIM + bcol];
    #pragma unroll
    for (int r = 0; r < 8; ++r)
      acc[r] += X[(tm * 16 + half * 8 + r) * D_DIM + k] * bv;
  }
#endif
  const float bias = bih[bcol];
  #pragma unroll
  for (int r = 0; r < 8; ++r) {
    const int row = tm * 16 + half * 8 + r;
    Z[row * G_DIM + bcol] = acc[r] + bias;
  }
}

// ---------------------------------------------------------------------------
// Kernel 4: sensory synapse precompute (parallel over t), packed b128 loads
// ---------------------------------------------------------------------------
__global__ void ncp_sensory(const float* __restrict__ X,
                            const float* __restrict__ input_w,
                            const float* __restrict__ input_b,
                            const float4* __restrict__ sensp,
                            float* __restrict__ wns,
                            float* __restrict__ wds) {
  __shared__ float inp[D_DIM];
  const int t = blockIdx.x;
  for (int d = threadIdx.x; d < D_DIM; d += blockDim.x)
    inp[d] = X[t * D_DIM + d] * input_w[d] + input_b[d];
  __syncthreads();

  const int u = threadIdx.x;  // blockDim == 128
  float sn = 0.0f, sd = 0.0f;
  #pragma unroll 4
  for (int d = 0; d < D_DIM; ++d) {
    const float4 p = sensp[d * U_DIM + u];      // {swp, mu, sigma, erev}
    const float s = p.x * sigmoidf_((inp[d] - p.y) * p.z);
    sn += s * p.w;
    sd += s;
  }
  wns[t * U_DIM + u] = sn;
  wds[t * U_DIM + u] = sd;
}

// ---------------------------------------------------------------------------
// Kernel 5: sequential scan — single workgroup (512 threads = 16 waves).
// CDNA5 320KB LDS: the full 256KB packed synapse table {wp,mu,sigma,erev}
// is staged in LDS once, so all 6*512 unfold passes never touch memory.
// ---------------------------------------------------------------------------
__global__ void __launch_bounds__(512, 1)
ncp_scan(const float* __restrict__ Z,
         const float* __restrict__ Whh,
         const float* __restrict__ wns,
         const float* __restrict__ wds,
         const float4* __restrict__ synp,
         const float* __restrict__ cmt,
         const float* __restrict__ gl,
         const float* __restrict__ vleak,
         const float* __restrict__ h0,
         const float* __restrict__ c0,
         const float* __restrict__ out_w,
         const float* __restrict__ out_b,
         float* __restrict__ out) {
  __shared__ float4 syn[U_DIM * U_DIM];          // 256 KB packed synapse table
  __shared__ float h[U_DIM], c[U_DIM], v[U_DIM], z[G_DIM];
  __shared__ float pn[4][U_DIM], pd[4][U_DIM];
  const int tid   = threadIdx.x;
  const int u     = tid & (U_DIM - 1);
  const int chunk = tid >> 7;                    // 0..3, 32 rows each

  // stage synapse table into LDS (one-time, 512 threads x 32 float4s)
  for (int i = tid; i < U_DIM * U_DIM; i += 512)
    syn[i] = synp[i];
  if (tid < U_DIM) { h[tid] = h0[tid]; c[tid] = c0[tid]; }
  __syncthreads();

  for (int t = 0; t < T_LEN; ++t) {
    // ---- z = Z[t,:] + h @ W_hh  (all 512 threads, one output column each)
    {
      float acc = Z[t * G_DIM + tid];
      #pragma unroll 4
      for (int k = 0; k < U_DIM; ++k)
        acc += h[k] * Whh[k * G_DIM + tid];
      z[tid] = acc;
    }
    __syncthreads();

    // ---- mixed-memory LSTM cell (forget-gate +1 bias)
    if (tid < U_DIM) {
      const float ia = z[tid];
      const float ig = z[U_DIM + tid];
      const float fg = z[2 * U_DIM + tid];
      const float og = z[3 * U_DIM + tid];
      const float cc = c[tid] * sigmoidf_(fg + 1.0f) + tanhf(ia) * sigmoidf_(ig);
      c[tid] = cc;
      v[tid] = tanhf(cc) * sigmoidf_(og);
    }
    __syncthreads();

    // ---- LTC ODE: 6 unfolds, all-LDS synapse reads (ds_load_b128)
    for (int it = 0; it < UNFOLDS; ++it) {
      float sn = 0.0f, sd = 0.0f;
      const int s0 = chunk * 32;
      #pragma unroll 4
      for (int s = s0; s < s0 + 32; ++s) {
        const float4 p = syn[s * U_DIM + u];     // {wp, mu, sigma, erev}
        const float a = p.x * sigmoidf_((v[s] - p.y) * p.z);
        sn += a * p.w;
        sd += a;
      }
      pn[chunk][u] = sn;
      pd[chunk][u] = sd;
      __syncthreads();
      if (tid < U_DIM) {
        const float num = cmt[u] * v[u] + gl[u] * vleak[u]
                        + pn[0][u] + pn[1][u] + pn[2][u] + pn[3][u]
                        + wns[t * U_DIM + u];
        const float den = cmt[u] + gl[u]
                        + pd[0][u] + pd[1][u] + pd[2][u] + pd[3][u]
                        + wds[t * U_DIM + u];
        v[u] = num / (den + EPS_LTC);
      }
      __syncthreads();
    }

    // ---- motor output + carry (h <- v for next LSTM step)
    if (tid < M_DIM) out[t * M_DIM + tid] = v[tid] * out_w[tid] + out_b[tid];
    if (tid < U_DIM) h[tid] = v[tid];
    __syncthreads();
  }
}

// ---------------------------------------------------------------------------
extern "C" void kernel_launch(void* const* d_in, const int* in_sizes, int n_in,
                              void* d_out, int out_size, void* d_ws, size_t ws_size,
                              hipStream_t stream) {
  const int*   sentence   = (const int*)  d_in[0];
  const float* embed      = (const float*)d_in[1];
  const float* W_ih       = (const float*)d_in[2];
  const float* b_ih       = (const float*)d_in[3];
  const float* W_hh       = (const float*)d_in[4];
  const float* gleak      = (const float*)d_in[5];
  const float* vleak      = (const float*)d_in[6];
  const float* cm         = (const float*)d_in[7];
  const float* sigma      = (const float*)d_in[8];
  const float* mu         = (const float*)d_in[9];
  const float* w          = (const float*)d_in[10];
  const float* erev       = (const float*)d_in[11];
  const float* sens_sigma = (const float*)d_in[12];
  const float* sens_mu    = (const float*)d_in[13];
  const float* sens_w     = (const float*)d_in[14];
  const float* sens_erev  = (const float*)d_in[15];
  const float* mask       = (const float*)d_in[16];
  const float* sens_mask  = (const float*)d_in[17];
  const float* input_w    = (const float*)d_in[18];
  const float* input_b    = (const float*)d_in[19];
  const float* output_w   = (const float*)d_in[20];
  const float* output_b   = (const float*)d_in[21];
  const float* h0         = (const float*)d_in[22];
  const float* c0         = (const float*)d_in[23];

  float* out = (float*)d_out;

  // workspace carve-up
  float*  Xg    = (float*)d_ws;               // [T, D]        1 MB
  float*  Zg    = Xg  + T_LEN * D_DIM;        // [T, 4U]       1 MB
  float*  wns   = Zg  + T_LEN * G_DIM;        // [T, U]        256 KB
  float*  wds   = wns + T_LEN * U_DIM;        // [T, U]        256 KB
  float*  cmt   = wds + T_LEN * U_DIM;        // [U]
  float*  glp   = cmt + U_DIM;                // [U]
  float*  pad   = glp + U_DIM;
  // align float4 region to 16 bytes
  uintptr_t p4  = (((uintptr_t)pad) + 15) & ~(uintptr_t)15;
  float4* sensp = (float4*)p4;                // [D, U] packed  1 MB
  float4* synp  = sensp + D_DIM * U_DIM;      // [U, U] packed  256 KB

  ncp_gather_embed<<<T_LEN, 256, 0, stream>>>(sentence, embed, Xg);

  ncp_precompute<<<(D_DIM * U_DIM + 255) / 256, 256, 0, stream>>>(
      sens_w, sens_mask, sens_mu, sens_sigma, sens_erev,
      w, mask, mu, sigma, erev, cm, gleak,
      sensp, synp, cmt, glp);

  // 1024 tiles, 8 waves (256 threads) per block -> 128 blocks
  ncp_gemm_wmma<<<128, 256, 0, stream>>>(Xg, W_ih, b_ih, Zg);

  ncp_sensory<<<T_LEN, 128, 0, stream>>>(Xg, input_w, input_b, sensp,
                                         wns, wds);

  ncp_scan<<<1, 512, 0, stream>>>(Zg, W_hh, wns, wds, synp,
                                  cmt, glp, vleak, h0, c0,
                                  output_w, output_b, out);
}